// DenormalJointNet_22462678958222
// MI455X (gfx1250) — compile-verified
//
#include <hip/hip_runtime.h>
#include <math.h>

typedef float v2f __attribute__((ext_vector_type(2)));
typedef float v8f __attribute__((ext_vector_type(8)));

#define B_ 4
#define T_ 512
#define U_ 64
#define V_ 1024

// ---------------------------------------------------------------------------
// Kernel 1: row-wise log_softmax for tn (B*T rows) and pn (B*U rows, col 0
// zeroed) into workspace. One 256-thread block per row, 4 floats/thread.
// ---------------------------------------------------------------------------
__global__ __launch_bounds__(256) void lsm_kernel(const float* __restrict__ tn,
                                                  const float* __restrict__ pn,
                                                  float* __restrict__ tn_ls,
                                                  float* __restrict__ pn_ls) {
  __shared__ float red[256];
  const int row   = blockIdx.x;
  const bool is_pn = row >= B_ * T_;
  const float* src = is_pn ? (pn + (size_t)(row - B_ * T_) * V_)
                           : (tn + (size_t)row * V_);
  float* dst = is_pn ? (pn_ls + (size_t)(row - B_ * T_) * V_)
                     : (tn_ls + (size_t)row * V_);
  const int tid = threadIdx.x;

  float4 x = reinterpret_cast<const float4*>(src)[tid];

  // block-wide max
  float m = fmaxf(fmaxf(x.x, x.y), fmaxf(x.z, x.w));
  red[tid] = m;
  __syncthreads();
  for (int s = 128; s > 0; s >>= 1) {
    if (tid < s) red[tid] = fmaxf(red[tid], red[tid + s]);
    __syncthreads();
  }
  m = red[0];
  __syncthreads();

  // block-wide sum of exp(x - m)
  float sum = expf(x.x - m) + expf(x.y - m) + expf(x.z - m) + expf(x.w - m);
  red[tid] = sum;
  __syncthreads();
  for (int s = 128; s > 0; s >>= 1) {
    if (tid < s) red[tid] += red[tid + s];
    __syncthreads();
  }
  const float lse = m + logf(red[0]);

  float4 y;
  y.x = x.x - lse;
  y.y = x.y - lse;
  y.z = x.z - lse;
  y.w = x.w - lse;
  if (is_pn && tid == 0) y.x = 0.0f;  // zero the <blk> column (v == 0)
  reinterpret_cast<float4*>(dst)[tid] = y;
}

// ---------------------------------------------------------------------------
// Kernel 2: broadcast joint via fp32 WMMA accumulate.
//   D[u,v] = A x B + C  with A = e_0 column of ones, B[0,:] = tn[t, v0:v0+16],
//   C = pn[u0:u0+16, v0:v0+16]  =>  D[u,v] = tn[t,v] + pn[u,v]   (exact fp32)
// Grid: (V/64, U/16, B*T/128). 128 threads = 4 waves; each wave owns a
// 16-wide v sub-tile, keeps its pn C-tile resident in 8 VGPRs for all 128 t.
// tn slice (128 x 64 floats, 32 KB) staged in LDS once per block.
// B-matrix LDS reads are UNPREDICATED: lanes 16-31 read a 64-float zero pad
// whose banks are disjoint from the data banks, so no EXEC save/restore and
// no per-iteration zero-movs in the inner loop.
// ---------------------------------------------------------------------------
#define ZPAD (128 * 64)  // float offset of the zero block in LDS

__global__ __launch_bounds__(128) void joint_wmma_kernel(
    const float* __restrict__ tn_ls, const float* __restrict__ pn_ls,
    float* __restrict__ out) {
  __shared__ float tn_lds[128 * 64 + 64];

  const int v0 = blockIdx.x * 64;
  const int u0 = blockIdx.y * 16;
  const int bz = blockIdx.z;
  const int b  = bz >> 2;
  const int t0 = (bz & 3) * 128;

  const int tid = threadIdx.x;

  // zero pad for lanes 16-31 of the B-matrix reads
  if (tid < 64) tn_lds[ZPAD + tid] = 0.0f;

  // stage tn[b, t0:t0+128, v0:v0+64] into LDS (float4, fully coalesced)
  {
    const float4* src =
        reinterpret_cast<const float4*>(tn_ls + ((size_t)(b * T_ + t0) * V_) + v0);
    float4* dstl = reinterpret_cast<float4*>(tn_lds);
#pragma unroll
    for (int i = 0; i < 16; ++i) {
      const int idx = tid + i * 128;       // 0 .. 2047 float4s
      const int tt  = idx >> 4;            // row (16 float4 per row)
      const int vv  = idx & 15;
      dstl[tt * 16 + vv] = src[(size_t)tt * (V_ / 4) + vv];
    }
  }
  __syncthreads();

  const int lane  = tid & 31;
  const int wv    = tid >> 5;  // wave id 0..3 -> v sub-tile
  const int lmod  = lane & 15;
  const int lhalf = lane >> 4;
  const int vw    = v0 + wv * 16;

  // A matrix (16x4): column k==0 all ones -> VGPR0 lanes 0-15 = 1.0, rest 0
  v2f a;
  a.x = (lhalf == 0) ? 1.0f : 0.0f;
  a.y = 0.0f;

  // C tile: pn_ls[b, u0 + i + lhalf*8, vw + lmod] per the 16x16 f32 layout
  v8f c;
  {
    const float* p = pn_ls + ((size_t)(b * U_ + u0 + lhalf * 8) * V_) + vw + lmod;
#pragma unroll
    for (int i = 0; i < 8; ++i) c[i] = p[(size_t)i * V_];
  }

  // per-lane output base: row u = u0 + i + lhalf*8, col v = vw + lmod
  float* obase =
      out + (((size_t)(b * T_ + t0) * U_ + u0 + lhalf * 8) * V_) + vw + lmod;

  // Unpredicated B-matrix source: lanes 0-15 walk the staged slice (banks
  // wv*16..wv*16+15), lanes 16-31 pin to the zero pad at disjoint banks
  // ((wv+1)*16 + lmod) & 63. Per-lane stride: 64 floats or 0.
  const int bidx = (lhalf == 0) ? (wv * 16 + lmod)
                                : (ZPAD + (((wv + 1) * 16 + lmod) & 63));
  const int binc = (lhalf == 0) ? 64 : 0;
  const float* bp = tn_lds + bidx;

  for (int t = 0; t < 128; ++t) {
    v2f bb;
    bb.x = *bp;      // ds_load_b32, all 32 lanes, no EXEC masking
    bb.y = 0.0f;     // K=1/K=3 rows: loop-invariant zero VGPR
    bp += binc;

    v8f d = __builtin_amdgcn_wmma_f32_16x16x4_f32(
        /*neg_a=*/false, a, /*neg_b=*/false, bb,
        /*c_mod=*/(short)0, c, /*reuse_a=*/false, /*reuse_b=*/false);

    float* o = obase + (size_t)t * (U_ * V_);
#pragma unroll
    for (int i = 0; i < 8; ++i)
      __builtin_nontemporal_store(d[i], o + (size_t)i * V_);  // streaming out
  }
}

// ---------------------------------------------------------------------------
extern "C" void kernel_launch(void* const* d_in, const int* in_sizes, int n_in,
                              void* d_out, int out_size, void* d_ws,
                              size_t ws_size, hipStream_t stream) {
  const float* tn = (const float*)d_in[0];  // (B, T, V)
  const float* pn = (const float*)d_in[1];  // (B, U, V)
  float* out = (float*)d_out;               // (B, T, U, V)

  float* tn_ls = (float*)d_ws;                       // B*T*V floats (8 MB)
  float* pn_ls = tn_ls + (size_t)B_ * T_ * V_;       // B*U*V floats (1 MB)

  lsm_kernel<<<dim3(B_ * T_ + B_ * U_), 256, 0, stream>>>(tn, pn, tn_ls, pn_ls);

  dim3 grid(V_ / 64, U_ / 16, B_ * (T_ / 128));  // (16, 4, 16)
  joint_wmma_kernel<<<grid, 128, 0, stream>>>(tn_ls, pn_ls, out);
}